// FlashAttentionTiled_80032420594143
// MI455X (gfx1250) — compile-verified
//
#include <hip/hip_runtime.h>
#include <math.h>

typedef float v2f __attribute__((ext_vector_type(2)));
typedef float v8f __attribute__((ext_vector_type(8)));

#define S_LEN     8192
#define DHEAD     128
#define OUTW      (DHEAD + 2)
#define MASK_FILL (-9984.0f)

#define WAVES_PER_WG 4
// waves 0-1: low strip rows [32b, 32b+32) ; waves 2-3: high strip rows [S-32(b+1), S-32b)
#define NUM_BLOCKS   (S_LEN / 64)

// padded LDS strides (all even -> 8B-aligned b64 fragment loads, spread banks)
#define KT_STR 130   // kTile[n][d]   n in [0,16), d in [0,128)
#define VT_STR 18    // vTile[d][k]   d in [0,128), k in [0,16)
#define PT_STR 18    // pScr[wave][m][k]

__launch_bounds__(WAVES_PER_WG * 32)
__global__ void fa_fwd_kernel(const float* __restrict__ q,
                              const float* __restrict__ k,
                              const float* __restrict__ v,
                              float* __restrict__ out)
{
    __shared__ __align__(16) float kTile[16 * KT_STR];
    __shared__ __align__(16) float vTile[DHEAD * VT_STR];
    __shared__ __align__(16) float pScr[WAVES_PER_WG * 16 * PT_STR];

    const int tid  = threadIdx.x;
    const int wave = tid >> 5;
    const int lane = tid & 31;
    const int half = lane >> 4;   // 0: rows 0-7 / K 0-1 ; 1: rows 8-15 / K 2-3
    const int l16  = lane & 15;

    // balanced causal pairing: low strip for waves 0-1, mirrored high strip for waves 2-3
    const int b     = blockIdx.x;
    const int qrow0 = (wave < 2) ? (b * 32 + wave * 16)
                                 : (S_LEN - (b + 1) * 32 + (wave - 2) * 16);
    const int myDiag = qrow0 >> 4;                       // this wave's diagonal key tile
    const int jend   = ((S_LEN - b * 32) >> 4) - 1;      // high strip's diagonal tile

    // ---- Q fragments: A-matrix layout for V_WMMA_F32_16X16X4_F32 ----
    // lane<16 : (K=4kk, 4kk+1) of row l16 ; lane>=16 : (K=4kk+2, 4kk+3)
    v2f qa[32];
    {
        const float* qrow = q + (size_t)(qrow0 + l16) * DHEAD + 2 * half;
        #pragma unroll
        for (int kk = 0; kk < 32; ++kk)
            qa[kk] = *(const v2f*)(qrow + 4 * kk);
    }

    v8f o[8];
    #pragma unroll
    for (int t = 0; t < 8; ++t) o[t] = v8f{};

    float mrow[8], lrow[8];
    #pragma unroll
    for (int r = 0; r < 8; ++r) { mrow[r] = -__builtin_inff(); lrow[r] = 0.0f; }

    float* pMine = pScr + wave * 16 * PT_STR;

    for (int j = 0; j <= jend; ++j) {
        // ---- cooperative stage of K tile: kTile[n][d] = k[d][16j+n] ----
        #pragma unroll
        for (int i = 0; i < 16; ++i) {
            int idx = i * 128 + tid;          // 0..2047
            int d = idx >> 4, n = idx & 15;   // coalesced 64B reads per 16 lanes
            kTile[n * KT_STR + d] = k[(size_t)d * S_LEN + (size_t)j * 16 + n];
        }
        // ---- cooperative stage of V tile: vTile[d][kk] = v[16j+kk][d] ----
        #pragma unroll
        for (int i = 0; i < 16; ++i) {
            vTile[tid * VT_STR + i] = v[(size_t)(j * 16 + i) * DHEAD + tid];
        }
        __syncthreads();

        if (j <= myDiag) {
            // ---- S = Q * K^T  (16x16 tile, K-dim 128 = 32 WMMA steps) ----
            v8f s0 = v8f{}, s1 = v8f{};
            #pragma unroll
            for (int kk = 0; kk < 32; kk += 2) {
                v2f b0 = *(const v2f*)(kTile + l16 * KT_STR + 4 * kk + 2 * half);
                v2f b1 = *(const v2f*)(kTile + l16 * KT_STR + 4 * (kk + 1) + 2 * half);
                s0 = __builtin_amdgcn_wmma_f32_16x16x4_f32(false, qa[kk],     false, b0,
                                                           (short)0, s0, false, false);
                s1 = __builtin_amdgcn_wmma_f32_16x16x4_f32(false, qa[kk + 1], false, b1,
                                                           (short)0, s1, false, false);
            }
            v8f sf = s0 + s1;

            // ---- causal mask (diagonal tile only): keep iff col<=row ----
            if (j == myDiag) {
                #pragma unroll
                for (int r = 0; r < 8; ++r) {
                    int M = r + 8 * half;
                    sf[r] = (l16 <= M) ? sf[r] : MASK_FILL;
                }
            }

            // ---- online softmax: row max / exp / row sum / rescale ----
            float scale[8];
            #pragma unroll
            for (int r = 0; r < 8; ++r) {
                float x = sf[r];
                x = fmaxf(x, __shfl_xor(x, 1));
                x = fmaxf(x, __shfl_xor(x, 2));
                x = fmaxf(x, __shfl_xor(x, 4));
                x = fmaxf(x, __shfl_xor(x, 8));
                float nm = fmaxf(mrow[r], x);
                float p  = __expf(sf[r] - nm);
                sf[r] = p;
                float s = p;
                s += __shfl_xor(s, 1);
                s += __shfl_xor(s, 2);
                s += __shfl_xor(s, 4);
                s += __shfl_xor(s, 8);
                scale[r] = __expf(mrow[r] - nm);
                lrow[r]  = lrow[r] * scale[r] + s;
                mrow[r]  = nm;
            }
            #pragma unroll
            for (int t = 0; t < 8; ++t) {
                #pragma unroll
                for (int r = 0; r < 8; ++r) o[t][r] *= scale[r];
            }

            // ---- C-layout -> A-layout for P via wave-private LDS scratch ----
            #pragma unroll
            for (int r = 0; r < 8; ++r)
                pMine[(r + 8 * half) * PT_STR + l16] = sf[r];
            asm volatile("s_wait_dscnt 0" ::: "memory"); // in-wave LDS RAW fence
            v2f pa[4];
            #pragma unroll
            for (int s4 = 0; s4 < 4; ++s4)
                pa[s4] = *(const v2f*)(pMine + l16 * PT_STR + 4 * s4 + 2 * half);

            // ---- O += P * V  (8 N-tiles x 4 K-steps of WMMA) ----
            #pragma unroll
            for (int t = 0; t < 8; ++t) {
                v8f acc = o[t];
                #pragma unroll
                for (int s4 = 0; s4 < 4; ++s4) {
                    v2f bb = *(const v2f*)(vTile + (16 * t + l16) * VT_STR + 4 * s4 + 2 * half);
                    acc = __builtin_amdgcn_wmma_f32_16x16x4_f32(false, pa[s4], false, bb,
                                                                (short)0, acc, false, false);
                }
                o[t] = acc;
            }
        }
        __syncthreads();
    }

    // ---- epilogue: unnormalized O, then logsumexp and row max ----
    const int rowbase = qrow0 + 8 * half;
    #pragma unroll
    for (int t = 0; t < 8; ++t) {
        #pragma unroll
        for (int r = 0; r < 8; ++r)
            out[(size_t)(rowbase + r) * OUTW + 16 * t + l16] = o[t][r];
    }
    if (l16 == 0) {
        #pragma unroll
        for (int r = 0; r < 8; ++r) {
            size_t row = (size_t)(rowbase + r);
            out[row * OUTW + DHEAD]     = logf(lrow[r]) + mrow[r]; // l
            out[row * OUTW + DHEAD + 1] = mrow[r];                 // m
        }
    }
}

extern "C" void kernel_launch(void* const* d_in, const int* in_sizes, int n_in,
                              void* d_out, int out_size, void* d_ws, size_t ws_size,
                              hipStream_t stream) {
    (void)in_sizes; (void)n_in; (void)out_size; (void)d_ws; (void)ws_size;
    const float* q = (const float*)d_in[0];
    const float* k = (const float*)d_in[1];   // (D, S) pre-transposed
    const float* v = (const float*)d_in[2];
    float* out = (float*)d_out;

    dim3 grid(NUM_BLOCKS);                    // 128 workgroups, balanced causal work
    dim3 block(WAVES_PER_WG * 32);            // 4 waves (wave32)
    hipLaunchKernelGGL(fa_fwd_kernel, grid, block, 0, stream, q, k, v, out);
}